// MotifEncoder_13529146982746
// MI455X (gfx1250) — compile-verified
//
#include <hip/hip_runtime.h>
#include <hip/hip_bf16.h>
#include <math.h>

// Problem constants (match reference)
#define H_   128
#define K_   8
#define E_   10
#define PN_  6
#define PE_  8
#define AF_  9
#define EF_  3

typedef float v2f __attribute__((ext_vector_type(2)));
typedef float v8f __attribute__((ext_vector_type(8)));

// ---------------------------------------------------------------------------
// Weight pair-packing: Wp[((k>>1)*128 + col)*2 + (k&1)] = W[k*128 + col].
// Makes a WMMA B-fragment (W[ka][col], W[ka+1][col]) one contiguous b64 load.
// ---------------------------------------------------------------------------
__global__ void pack_weights_kernel(const float* __restrict__ W,
                                    float* __restrict__ Wp, int total)
{
  const int idx = blockIdx.x * blockDim.x + threadIdx.x;
  if (idx >= total) return;
  const int col = idx & (H_ - 1);
  const int k   = idx >> 7;
  Wp[((((k >> 1) * H_) + col) << 1) + (k & 1)] = W[idx];
}

// ---------------------------------------------------------------------------
// Core GEMM micro-kernel: one 16-wide column tile, NRT row tiles, via
// V_WMMA_F32_16X16X4_F32.
//   A: LDS row-major (8B-aligned; stride = 4 mod 8 -> conflict-free column
//      reads).  Per-lane base hoisted; k-offset is affine -> DS imm offsets.
//   B: pair-packed weights. Address = lane-invariant base + k0*512B, so LSR
//      emits one base register + immediate offsets (no per-load v_lshl_add_u64
//      / s_wait_xcnt churn).  One global_load_b64 shared by NRT WMMAs.
// unroll capped at 4 to bound in-flight loads (no VGPR spills).
// ---------------------------------------------------------------------------
template<int NRT>
__device__ __forceinline__ void gemm_coltile(const float* lA, int lda,
                                             const float* gBp, int kdim,
                                             float bias, int lane, v8f* acc)
{
  const int rn = lane & 15;   // A row (M) within tile / B col (N)
  const int hi = lane >> 4;   // which K-pair of the 4-wide step
  const float* bp = gBp + (hi * H_ + rn) * 2;  // + k0*H_ floats per k-step
  const float* ap = lA + rn * lda + 2 * hi;    // + rt*16*lda + k0
  #pragma unroll
  for (int rt = 0; rt < NRT; ++rt) {
    #pragma unroll
    for (int j = 0; j < 8; ++j) acc[rt][j] = bias;
  }
  #pragma unroll 4
  for (int k0 = 0; k0 < kdim; k0 += 4) {
    const v2f b = *(const v2f*)(bp + (size_t)k0 * H_);
    #pragma unroll
    for (int rt = 0; rt < NRT; ++rt) {
      const v2f a = *(const v2f*)(ap + rt * 16 * lda + k0);
      acc[rt] = __builtin_amdgcn_wmma_f32_16x16x4_f32(false, a, false, b,
                                                      (short)0, acc[rt],
                                                      false, false);
    }
  }
}

// ---------------------------------------------------------------------------
// h0[row] = x[motif_nodes[row]] @ atom_W + atom_b   (4 rows/block, 512 thr)
// ---------------------------------------------------------------------------
__global__ void embed_nodes_kernel(const float* __restrict__ x,
                                   const int* __restrict__ motif_nodes,
                                   const float* __restrict__ atom_W,
                                   const float* __restrict__ atom_b,
                                   float* __restrict__ h)
{
  __shared__ float xr[4][AF_];
  const int g = threadIdx.x >> 7, c = threadIdx.x & 127;
  const int row = blockIdx.x * 4 + g;
  const int node = motif_nodes[row];
  if (c < AF_) xr[g][c] = x[(size_t)node * AF_ + c];
  __syncthreads();
  float acc = atom_b[c];
  #pragma unroll
  for (int f = 0; f < AF_; ++f) acc += xr[g][f] * atom_W[f * H_ + c];
  h[(size_t)row * H_ + c] = acc;
}

__global__ void embed_edges_kernel(const float* __restrict__ eag,
                                   const int* __restrict__ eidx,
                                   const float* __restrict__ edge_W,
                                   const float* __restrict__ edge_b,
                                   float* __restrict__ ea)
{
  __shared__ float er[4][EF_];
  const int g = threadIdx.x >> 7, c = threadIdx.x & 127;
  const int row = blockIdx.x * 4 + g;
  const int e = eidx[row];
  if (c < EF_) er[g][c] = eag[(size_t)e * EF_ + c];
  __syncthreads();
  float acc = edge_b[c];
  #pragma unroll
  for (int f = 0; f < EF_; ++f) acc += er[g][f] * edge_W[f * H_ + c];
  ea[(size_t)row * H_ + c] = acc;
}

// ---------------------------------------------------------------------------
// One GIN-conv layer over 8 motifs (64 rows = 4 row tiles) per 256-thread
// block (8 waves; each wave owns one 16-wide column tile -> acc[4] = 32 VGPR).
// Phase 1 (column-owned within each 128-thread group; barrier-free):
//   z = (1+eps)*h + scatter_dst(ea * h[src])
// GEMM1 results stay in registers so zs can be reused as the relu buffer.
// h updated in place (motif graphs are closed under K).
// ---------------------------------------------------------------------------
__global__ void __launch_bounds__(256)
conv_layer_kernel(float* __restrict__ h,
                  const float* __restrict__ ea,
                  const int* __restrict__ esrc,
                  const int* __restrict__ edst,
                  const float* __restrict__ W1p,
                  const float* __restrict__ b1,
                  const float* __restrict__ W2p,
                  const float* __restrict__ b2,
                  const float* __restrict__ eps_arr, int layer)
{
  __shared__ float zs[64 * 132];        // z tile, then relu(z@W1+b1)
  __shared__ float hstage[2][8 * 132];  // raw-h slab per thread-group
  const int m0 = blockIdx.x * 8;
  const int tid = threadIdx.x, lane = tid & 31, wv = tid >> 5;
  const int g = tid >> 7, c = tid & 127;
  const float eps1 = 1.0f + eps_arr[layer];

  for (int mi = g; mi < 8; mi += 2) {   // group g owns motifs mi%2==g
    const int m = m0 + mi;
    float* hst = hstage[g];
    #pragma unroll
    for (int k = 0; k < K_; ++k) {
      const float v = h[(size_t)(m * K_ + k) * H_ + c];
      hst[k * 132 + c] = v;
      zs[(mi * K_ + k) * 132 + c] = eps1 * v;
    }
    #pragma unroll
    for (int e = 0; e < E_; ++e) {
      const int s = esrc[m * E_ + e];
      const int d = edst[m * E_ + e];
      const float eav = ea[(size_t)(m * E_ + e) * H_ + c];
      zs[(mi * K_ + d) * 132 + c] += eav * hst[s * 132 + c];
    }
  }
  __syncthreads();

  const int col = lane & 15, hi = lane >> 4;
  const float* W1l = W1p + (size_t)layer * H_ * H_;
  const float* b1l = b1 + layer * H_;
  v8f acc[4];
  gemm_coltile<4>(zs, 132, W1l + wv * 32, H_, b1l[wv * 16 + col], lane, acc);
  __syncthreads();                      // all zs reads done -> reuse as act buf
  #pragma unroll
  for (int rt = 0; rt < 4; ++rt) {
    #pragma unroll
    for (int j = 0; j < 8; ++j)
      zs[(rt * 16 + j + 8 * hi) * 132 + wv * 16 + col] = fmaxf(acc[rt][j], 0.0f);
  }
  __syncthreads();

  const float* W2l = W2p + (size_t)layer * H_ * H_;
  const float* b2l = b2 + layer * H_;
  gemm_coltile<4>(zs, 132, W2l + wv * 32, H_, b2l[wv * 16 + col], lane, acc);
  #pragma unroll
  for (int rt = 0; rt < 4; ++rt) {
    #pragma unroll
    for (int j = 0; j < 8; ++j)
      h[(size_t)(m0 * K_ + rt * 16 + j + 8 * hi) * H_ + wv * 16 + col] = acc[rt][j];
  }
}

// ---------------------------------------------------------------------------
// Attention pooling (2 motifs / 128 threads; no GEMM).
// ---------------------------------------------------------------------------
__global__ void attn_kernel(const float* __restrict__ h,
                            const float* __restrict__ ea,
                            const int* __restrict__ esrc,
                            const int* __restrict__ edst,
                            const float* __restrict__ attn_W,
                            const float* __restrict__ attn_b,
                            float* __restrict__ h_nodes)
{
  __shared__ float hs[16 * 132];
  __shared__ float eg[16 * 132];
  __shared__ float red[16 * 8];
  __shared__ float alph[16];
  const int m0 = blockIdx.x * 2, tid = threadIdx.x;

  for (int r = 0; r < 16; ++r) {
    hs[r * 132 + tid] = h[(size_t)(m0 * K_ + r) * H_ + tid];
    eg[r * 132 + tid] = 0.0f;
  }
  __syncthreads();
  for (int mi = 0; mi < 2; ++mi) {
    const int m = m0 + mi;
    #pragma unroll
    for (int e = 0; e < E_; ++e) {
      const int s = esrc[m * E_ + e];
      const int d = edst[m * E_ + e];
      const float v = ea[(size_t)(m * E_ + e) * H_ + tid];
      eg[(mi * K_ + s) * 132 + tid] += v;
      eg[(mi * K_ + d) * 132 + tid] += v;
    }
  }
  __syncthreads();

  const int row = tid >> 3, part = tid & 7;
  float p = 0.0f;
  #pragma unroll
  for (int i = 0; i < 16; ++i) {
    const int cc = part * 16 + i;
    p += hs[row * 132 + cc] * attn_W[cc] + eg[row * 132 + cc] * attn_W[H_ + cc];
  }
  red[row * 8 + part] = p;
  __syncthreads();
  if (part == 0) {
    float s = attn_b[0];
    #pragma unroll
    for (int i = 0; i < 8; ++i) s += red[row * 8 + i];
    alph[row] = 1.0f / (1.0f + __expf(-s));
  }
  __syncthreads();

  float o0 = 0.0f, o1 = 0.0f;
  #pragma unroll
  for (int k = 0; k < K_; ++k) {
    o0 += alph[k] * hs[k * 132 + tid];
    o1 += alph[K_ + k] * hs[(K_ + k) * 132 + tid];
  }
  h_nodes[(size_t)m0 * H_ + tid] = o0;
  h_nodes[(size_t)(m0 + 1) * H_ + tid] = o1;
}

// ---------------------------------------------------------------------------
// Edge MLP: 2 motifs per 256-thread block. A = [h[src], h[dst], ea]
// (20 rows padded to 32). Each wave owns one col tile, NRT=2 (acc = 16 VGPR).
// relu(A @ emlp_W + b), mean over E, locally (no atomics).
// ---------------------------------------------------------------------------
__global__ void __launch_bounds__(256)
emlp_kernel(const float* __restrict__ h,
            const float* __restrict__ ea,
            const int* __restrict__ esrc,
            const int* __restrict__ edst,
            const float* __restrict__ emlp_Wp,
            const float* __restrict__ emlp_b,
            float* __restrict__ h_edges)
{
  __shared__ float hs[16 * 132];
  __shared__ float A[32 * 388];         // reused as O[32*132] after the GEMM
  const int m0 = blockIdx.x * 2, tid = threadIdx.x;
  const int lane = tid & 31, wv = tid >> 5;
  const int g = tid >> 7, c = tid & 127;

  #pragma unroll
  for (int r8 = 0; r8 < 8; ++r8) {      // group g loads its motif's 8 rows
    const int r = g * 8 + r8;
    hs[r * 132 + c] = h[(size_t)(m0 * K_ + r) * H_ + c];
  }
  for (int r = 20 + g; r < 32; r += 2) {
    A[r * 388 + c] = 0.0f;
    A[r * 388 + 128 + c] = 0.0f;
    A[r * 388 + 256 + c] = 0.0f;
  }
  // group g builds its motif's 10 rows (reads only rows it wrote; no barrier)
  {
    const int m = m0 + g;
    #pragma unroll
    for (int e = 0; e < E_; ++e) {
      const int r = g * E_ + e;
      const int s = esrc[m * E_ + e];
      const int d = edst[m * E_ + e];
      A[r * 388 + c]       = hs[(g * K_ + s) * 132 + c];
      A[r * 388 + 128 + c] = hs[(g * K_ + d) * 132 + c];
      A[r * 388 + 256 + c] = ea[(size_t)(m * E_ + e) * H_ + c];
    }
  }
  __syncthreads();

  const int col = lane & 15, hi = lane >> 4;
  v8f acc[2];
  gemm_coltile<2>(A, 388, emlp_Wp + wv * 32, 384, emlp_b[wv * 16 + col], lane, acc);
  __syncthreads();                      // all A reads done; reuse A as O
  float* O = A;
  #pragma unroll
  for (int rt = 0; rt < 2; ++rt) {
    #pragma unroll
    for (int j = 0; j < 8; ++j)
      O[(rt * 16 + j + 8 * hi) * 132 + wv * 16 + col] = fmaxf(acc[rt][j], 0.0f);
  }
  __syncthreads();

  float s = 0.0f;
  #pragma unroll
  for (int e = 0; e < E_; ++e) s += O[(g * E_ + e) * 132 + c];
  h_edges[(size_t)(m0 + g) * H_ + c] = s * (1.0f / E_);
}

// ---------------------------------------------------------------------------
// motifs_X = [h_nodes + h_edges, tnode] @ motif_W + motif_b   (K=256 GEMM)
// 16 motifs per 256-thread block; one thread per A column.
// ---------------------------------------------------------------------------
__global__ void __launch_bounds__(256)
motif_out_kernel(const float* __restrict__ h_nodes,
                 const float* __restrict__ h_edges,
                 const int* __restrict__ motif_types,
                 const float* __restrict__ type_emb,
                 const float* __restrict__ motif_Wp,
                 const float* __restrict__ motif_b,
                 float n_nodes, float* __restrict__ out)
{
  __shared__ float A[16 * 260];
  const int m0 = blockIdx.x * 16, tid = threadIdx.x;
  const int lane = tid & 31, wv = tid >> 5;

  for (int r = 0; r < 16; ++r) {
    const int m = m0 + r;
    if (tid < H_) {
      A[r * 260 + tid] = h_nodes[(size_t)m * H_ + tid] + h_edges[(size_t)m * H_ + tid];
    } else {
      const int cc = tid - H_;
      const int ty = motif_types[m];
      float te = type_emb[ty * H_ + cc];
      if (cc >= H_ / 2) te *= n_nodes;
      A[r * 260 + H_ + cc] = te;
    }
  }
  __syncthreads();
  const int col = lane & 15, hi = lane >> 4;
  v8f acc[1];
  gemm_coltile<1>(A, 260, motif_Wp + wv * 32, 2 * H_, motif_b[wv * 16 + col], lane, acc);
  #pragma unroll
  for (int j = 0; j < 8; ++j)
    out[(size_t)(m0 + j + 8 * hi) * H_ + wv * 16 + col] = acc[0][j];
}

// ---------------------------------------------------------------------------
// Motif-edge attrs, 32 rows per 256-thread block (2 row tiles). Linearity:
// h_atom_edge = (sum x)@atom_W + 6b_a + (sum ea)@edge_W + 8b_e.
// out = [type_emb[st]+type_emb[en], h_atom_edge] @ me_W + me_b   (K=256)
// ---------------------------------------------------------------------------
__global__ void __launch_bounds__(256)
medge_kernel(const float* __restrict__ x,
             const float* __restrict__ eag,
             const int* __restrict__ attr_nodes,
             const int* __restrict__ attr_edges,
             const int* __restrict__ mei,
             const int* __restrict__ motif_types,
             const float* __restrict__ type_emb,
             const float* __restrict__ atom_W,
             const float* __restrict__ atom_b,
             const float* __restrict__ edge_W,
             const float* __restrict__ edge_b,
             const float* __restrict__ me_Wp,
             const float* __restrict__ me_b,
             int ME_, float* __restrict__ out)
{
  __shared__ float A[32 * 260];
  __shared__ float sx[32 * AF_];
  __shared__ float se[32 * EF_];
  const int r0 = blockIdx.x * 32, tid = threadIdx.x;
  const int lane = tid & 31, wv = tid >> 5;

  for (int idx = tid; idx < 32 * AF_; idx += 256) {
    const int r = idx / AF_, f = idx % AF_;
    const int* an = attr_nodes + (size_t)(r0 + r) * PN_;
    float s = 0.0f;
    #pragma unroll
    for (int i = 0; i < PN_; ++i) s += x[(size_t)an[i] * AF_ + f];
    sx[idx] = s;
  }
  for (int idx = tid; idx < 32 * EF_; idx += 256) {
    const int r = idx / EF_, f = idx % EF_;
    const int* ae = attr_edges + (size_t)(r0 + r) * PE_;
    float s = 0.0f;
    #pragma unroll
    for (int i = 0; i < PE_; ++i) s += eag[(size_t)ae[i] * EF_ + f];
    se[idx] = s;
  }
  __syncthreads();

  if (tid < H_) {                       // waves 0-3: t_edge half
    for (int r = 0; r < 32; ++r) {
      const int e = r0 + r;
      const int st = motif_types[mei[e]];
      const int en = motif_types[mei[ME_ + e]];
      A[r * 260 + tid] = type_emb[st * H_ + tid] + type_emb[en * H_ + tid];
    }
  } else {                              // waves 4-7: h_atom_edge half
    const int cc = tid - H_;
    const float bsum = (float)PN_ * atom_b[cc] + (float)PE_ * edge_b[cc];
    for (int r = 0; r < 32; ++r) {
      float hv = bsum;
      #pragma unroll
      for (int f = 0; f < AF_; ++f) hv += sx[r * AF_ + f] * atom_W[f * H_ + cc];
      #pragma unroll
      for (int f = 0; f < EF_; ++f) hv += se[r * EF_ + f] * edge_W[f * H_ + cc];
      A[r * 260 + H_ + cc] = hv;
    }
  }
  __syncthreads();

  const int col = lane & 15, hi = lane >> 4;
  v8f acc[2];
  gemm_coltile<2>(A, 260, me_Wp + wv * 32, 2 * H_, me_b[wv * 16 + col], lane, acc);
  #pragma unroll
  for (int rt = 0; rt < 2; ++rt) {
    #pragma unroll
    for (int j = 0; j < 8; ++j)
      out[(size_t)(r0 + rt * 16 + j + 8 * hi) * H_ + wv * 16 + col] = acc[rt][j];
  }
}

// ---------------------------------------------------------------------------
extern "C" void kernel_launch(void* const* d_in, const int* in_sizes, int n_in,
                              void* d_out, int out_size, void* d_ws, size_t ws_size,
                              hipStream_t stream)
{
  const float* x        = (const float*)d_in[0];
  const float* eag      = (const float*)d_in[1];
  const int*   mnodes   = (const int*)d_in[2];
  const int*   mtypes   = (const int*)d_in[3];
  const int*   esrc     = (const int*)d_in[4];
  const int*   edst     = (const int*)d_in[5];
  const int*   eeidx    = (const int*)d_in[6];
  const int*   mei      = (const int*)d_in[7];
  const int*   an       = (const int*)d_in[8];
  const int*   ae       = (const int*)d_in[9];
  const float* type_emb = (const float*)d_in[10];
  const float* atom_W   = (const float*)d_in[11];
  const float* atom_b   = (const float*)d_in[12];
  const float* edge_W   = (const float*)d_in[13];
  const float* edge_b   = (const float*)d_in[14];
  const float* conv_W1  = (const float*)d_in[15];
  const float* conv_b1  = (const float*)d_in[16];
  const float* conv_W2  = (const float*)d_in[17];
  const float* conv_b2  = (const float*)d_in[18];
  const float* conv_eps = (const float*)d_in[19];
  const float* attn_W   = (const float*)d_in[20];
  const float* attn_b   = (const float*)d_in[21];
  const float* emlp_W   = (const float*)d_in[22];
  const float* emlp_b   = (const float*)d_in[23];
  const float* motif_W  = (const float*)d_in[24];
  const float* motif_b  = (const float*)d_in[25];
  const float* me_W     = (const float*)d_in[26];
  const float* me_b     = (const float*)d_in[27];

  const int M  = in_sizes[3];           // 50000
  const int ME = in_sizes[8] / PN_;     // 100000
  const float n_nodes = (float)(in_sizes[0] / AF_);

  // Workspace: h[M*K*H] | ea[M*E*H] | h_nodes[M*H] | h_edges[M*H] | packed W
  float* h        = (float*)d_ws;
  float* ea       = h + (size_t)M * K_ * H_;
  float* h_nodes  = ea + (size_t)M * E_ * H_;
  float* h_edges  = h_nodes + (size_t)M * H_;
  float* W1p      = h_edges + (size_t)M * H_;
  float* W2p      = W1p + 2 * H_ * H_;
  float* emlp_Wp  = W2p + 2 * H_ * H_;
  float* motif_Wp = emlp_Wp + 384 * H_;
  float* me_Wp    = motif_Wp + 2 * H_ * H_;

  float* outX = (float*)d_out;
  float* outE = outX + (size_t)M * H_;

  // One-time weight packing (overwrites with identical values per call)
  pack_weights_kernel<<<(2 * H_ * H_) / 256, 256, 0, stream>>>(conv_W1, W1p, 2 * H_ * H_);
  pack_weights_kernel<<<(2 * H_ * H_) / 256, 256, 0, stream>>>(conv_W2, W2p, 2 * H_ * H_);
  pack_weights_kernel<<<(384 * H_) / 256, 256, 0, stream>>>(emlp_W, emlp_Wp, 384 * H_);
  pack_weights_kernel<<<(2 * H_ * H_) / 256, 256, 0, stream>>>(motif_W, motif_Wp, 2 * H_ * H_);
  pack_weights_kernel<<<(2 * H_ * H_) / 256, 256, 0, stream>>>(me_W, me_Wp, 2 * H_ * H_);

  embed_nodes_kernel<<<M * K_ / 4, 512, 0, stream>>>(x, mnodes, atom_W, atom_b, h);
  embed_edges_kernel<<<M * E_ / 4, 512, 0, stream>>>(eag, eeidx, edge_W, edge_b, ea);
  conv_layer_kernel<<<M / 8, 256, 0, stream>>>(h, ea, esrc, edst, W1p, conv_b1,
                                               W2p, conv_b2, conv_eps, 0);
  conv_layer_kernel<<<M / 8, 256, 0, stream>>>(h, ea, esrc, edst, W1p, conv_b1,
                                               W2p, conv_b2, conv_eps, 1);
  attn_kernel<<<M / 2, 128, 0, stream>>>(h, ea, esrc, edst, attn_W, attn_b, h_nodes);
  emlp_kernel<<<M / 2, 256, 0, stream>>>(h, ea, esrc, edst, emlp_Wp, emlp_b, h_edges);
  motif_out_kernel<<<M / 16, 256, 0, stream>>>(h_nodes, h_edges, mtypes, type_emb,
                                               motif_Wp, motif_b, n_nodes, outX);
  medge_kernel<<<ME / 32, 256, 0, stream>>>(x, eag, an, ae, mei, mtypes, type_emb,
                                            atom_W, atom_b, edge_W, edge_b,
                                            me_Wp, me_b, ME, outE);
}